// ContourFeatureMapExtractor_8804682957283
// MI455X (gfx1250) — compile-verified
//
#include <hip/hip_runtime.h>
#include <cstdint>
#include <cstddef>

// ---------------- problem constants (from reference) ----------------
#define B_    16
#define NL_   200
#define CF_   64
#define HW_   256
#define LL_   64
#define PLANE (HW_*HW_)      // 65536
#define HBINS 32768          // 32^3

// flat output offsets (reference return order, all float32)
#define O_NI   0u            // normals_in_image   (B,NL,2)   6400
#define O_CI   6400u         // centers_in_image   (B,NL,2)   6400
#define O_CB   12800u        // centers_in_body    (B,NL,3)   9600
#define O_PF   22400u        // pf                 (B,NL,LL)  204800
#define O_PB   227200u       // pb                 (B,NL,LL)  204800
#define O_VD   432000u       // valid_data_line    (B,NL)     3200
#define O_AMP  435200u       // lines_amplitude    (B,NL)     3200
#define O_SLP  438400u       // lines_slope        (B,NL)     3200
#define O_LF   441600u       // lines_feature      (B,CF,NL,LL) 13107200

typedef __attribute__((ext_vector_type(2))) float v2f;
typedef __attribute__((ext_vector_type(8))) float v8f;

// =====================================================================
// Kernel 1: pose transform (WMMA f32 16x16x4), projection, validity.
// One wave (32 threads) per (16-line tile, batch). 13 x 16 blocks.
//   centers_in_view[n,i] = sum_j A[n,j]*B[j,i],  A=[cx cy cz 1], B=[R^T; t]
// A layout (ISA 7.12.2, 32-bit A 16x4): VGPR0 = K0 (lanes 0-15) / K2 (16-31),
//                                       VGPR1 = K1 / K3.
// B layout (4x16, row striped across lanes, rows r & r+2 per VGPR):
//   VGPR0 = K0 (lanes 0-15) / K2 (16-31), VGPR1 = K1 / K3.
// D layout (16x16 f32): VGPR r = row r (lanes 0-15) / row r+8 (16-31).
// =====================================================================
__global__ __launch_bounds__(32)
void k1_pose(const float* __restrict__ tv,    // template_view (B,NL,8)
             const float* __restrict__ pose,  // (B,12) R row-major + t
             const float* __restrict__ cam,   // (B,6) w h fx fy cx cy
             float* __restrict__ out) {
  __shared__ float ldsC[16 * 16];
  __shared__ float ldsN[16 * 16];

  const int lane = threadIdx.x;   // 0..31
  const int tile = blockIdx.x;    // 0..12
  const int b    = blockIdx.y;    // 0..15
  const int m    = lane & 15;     // row / col within tile
  const int half = lane >> 4;     // 0: K{0,1}, 1: K{2,3}

  int nA = tile * 16 + m;
  if (nA > NL_ - 1) nA = NL_ - 1;             // clamp pad rows (tile 12)
  const float* tva = tv + ((size_t)b * NL_ + nA) * 8;
  const float* P   = pose + b * 12;

  // ---- A matrices: centers [cx cy cz 1], normals [nx ny nz 0]
  v2f aC, aN;
  if (half == 0) { aC.x = tva[0]; aC.y = tva[1]; aN.x = tva[3]; aN.y = tva[4]; }
  else           { aC.x = tva[2]; aC.y = 1.0f;   aN.x = tva[5]; aN.y = 0.0f;  }

  // ---- B matrices: col i = (R[i][0], R[i][1], R[i][2], t[i]); cols 3..15 = 0
  v2f bC; bC.x = 0.0f; bC.y = 0.0f;
  if (m < 3) {
    if (half == 0) { bC.x = P[m * 3 + 0]; bC.y = P[m * 3 + 1]; }  // K0,K1
    else           { bC.x = P[m * 3 + 2]; bC.y = P[9 + m];     }  // K2,K3=t
  }
  v2f bN = bC;
  if (half == 1) bN.y = 0.0f;                                     // no t for normals

#if __has_builtin(__builtin_amdgcn_wmma_f32_16x16x4_f32)
  v8f c0 = {0.f, 0.f, 0.f, 0.f, 0.f, 0.f, 0.f, 0.f};
  // full EXEC here: no early return / divergence wraps these calls
  v8f dC = __builtin_amdgcn_wmma_f32_16x16x4_f32(false, aC, false, bC,
                                                 (short)0, c0, false, false);
  v8f dN = __builtin_amdgcn_wmma_f32_16x16x4_f32(false, aN, false, bN,
                                                 (short)0, c0, false, false);
#pragma unroll
  for (int r = 0; r < 8; ++r) {               // unstripe D -> LDS 16x16
    const int row = r + half * 8;
    ldsC[row * 16 + m] = dC[r];
    ldsN[row * 16 + m] = dN[r];
  }
#else
  if (half == 0) {                            // scalar fallback
    const float cx = tva[0], cy = tva[1], cz = tva[2];
    const float nx = tva[3], ny = tva[4], nz = tva[5];
    for (int i = 0; i < 3; ++i) {
      ldsC[m * 16 + i] = P[i*3]*cx + P[i*3+1]*cy + P[i*3+2]*cz + P[9+i];
      ldsN[m * 16 + i] = P[i*3]*nx + P[i*3+1]*ny + P[i*3+2]*nz;
    }
  }
#endif
  __syncthreads();

  // ---- per-line epilogue (lanes 0..15, one line each)
  if (lane < 16) {
    const int n = tile * 16 + lane;
    if (n < NL_) {
      const int bn = b * NL_ + n;
      const float* t8 = tv + (size_t)bn * 8;
      const float cvx = ldsC[lane*16+0], cvy = ldsC[lane*16+1], cvz = ldsC[lane*16+2];
      const float nvx = ldsN[lane*16+0], nvy = ldsN[lane*16+1];
      const float* cm = cam + b * 6;
      const float sw = cm[0], sh = cm[1], fx = cm[2], fy = cm[3], ppx = cm[4], ppy = cm[5];

      const float z  = cvz;
      const float zs = fmaxf(z, 1e-4f);
      const float cix = (cvx / zs) * fx + ppx;
      const float ciy = (cvy / zs) * fy + ppy;
      bool cvalid = (z > 1e-4f) && cix >= 0.f && cix <= sw - 1.f
                                && ciy >= 0.f && ciy <= sh - 1.f;

      const float den = fmaxf(sqrtf(nvx*nvx + nvy*nvy), 1e-12f);
      const float nix = nvx / den, niy = nvy / den;

      const float cont = fminf(t8[6] * fx / z, t8[7] * fx / z);
      bool vdl = (cont >= 6.0f) && cvalid;
      const float p0x = cix - 31.5f * nix, p0y = ciy - 31.5f * niy;
      const float p1x = cix + 31.5f * nix, p1y = ciy + 31.5f * niy;
      vdl = vdl && p0x >= 0.f && p0x < (float)HW_ && p0y >= 0.f && p0y < (float)HW_
                && p1x >= 0.f && p1x < (float)HW_ && p1y >= 0.f && p1y < (float)HW_;

      out[O_NI + bn*2+0] = nix;   out[O_NI + bn*2+1] = niy;
      out[O_CI + bn*2+0] = cix;   out[O_CI + bn*2+1] = ciy;
      out[O_CB + bn*3+0] = t8[0]; out[O_CB + bn*3+1] = t8[1]; out[O_CB + bn*3+2] = t8[2];
      out[O_VD  + bn] = vdl ? 1.0f : 0.0f;
      out[O_AMP + bn] = 0.43f;
      out[O_SLP + bn] = 0.5f;
    }
  }
}

// =====================================================================
// Kernel 2: nearest-sample image -> histogram lookups -> pf/pb.
// One thread per (b,n,l). grid sample coord = p - 0.5 (exact unnorm math),
// jnp.round == rintf (round-half-even under default RTE).
// =====================================================================
__global__ __launch_bounds__(256)
void k2_hist(const float* __restrict__ image,  // (B,3,H,W)
             const float* __restrict__ fore,   // (B,32768)
             const float* __restrict__ back,   // (B,32768)
             float* __restrict__ out) {
  const int idx = blockIdx.x * blockDim.x + threadIdx.x;
  if (idx >= B_ * NL_ * LL_) return;
  const int l  = idx & (LL_ - 1);
  const int bn = idx >> 6;
  const int b  = bn / NL_;

  const float cx = out[O_CI + bn*2+0], cy = out[O_CI + bn*2+1];
  const float nx = out[O_NI + bn*2+0], ny = out[O_NI + bn*2+1];
  const float s  = (float)l - 31.5f;
  const float x  = cx + s * nx - 0.5f;
  const float y  = cy + s * ny - 0.5f;
  const int xi = (int)rintf(x), yi = (int)rintf(y);
  const bool valid = xi >= 0 && xi < HW_ && yi >= 0 && yi < HW_;
  const int xc = min(max(xi, 0), HW_ - 1), yc = min(max(yi, 0), HW_ - 1);

  const float* img = image + (size_t)b * 3 * PLANE + yc * HW_ + xc;
  float v0 = 0.f, v1 = 0.f, v2 = 0.f;
  if (valid) { v0 = img[0]; v1 = img[PLANE]; v2 = img[2 * PLANE]; }

  const int b0 = min(max((int)(v0 * 32.f), 0), 31);
  const int b1 = min(max((int)(v1 * 32.f), 0), 31);
  const int b2 = min(max((int)(v2 * 32.f), 0), 31);
  const int h  = (b0 * 32 + b1) * 32 + b2;

  float pf = fore[(size_t)b * HBINS + h];
  float pb = back[(size_t)b * HBINS + h];
  if (pf < 1e-5f && pb < 1e-5f) { pf = 1e-5f; pb = 1e-5f; }
  const float inv = 1.0f / (pf + pb);
  out[O_PF + idx] = pf * inv;
  out[O_PB + idx] = pb * inv;
}

// =====================================================================
// Kernel 3: bilinear feature sampling (the memory-bound bulk).
// One 256-thread block per line (b,n): stage 64 points' corner indices +
// weights in LDS once, then 16 iterations cover all 64 channels with
// coalesced gathers/stores. global_prefetch_b8 pipelines the next plane.
// =====================================================================
__global__ __launch_bounds__(256)
void k3_feat(const float* __restrict__ feature,  // (B,CF,H,W)
             float* __restrict__ out) {
  __shared__ int   sX0[LL_], sY0[LL_];
  __shared__ float sWX[LL_], sWY[LL_];

  const int bn = blockIdx.x;            // b*NL_+n
  const int b  = bn / NL_;
  const int n  = bn - b * NL_;
  const int tid = threadIdx.x;

  if (tid < LL_) {
    const float cx = out[O_CI + bn*2+0], cy = out[O_CI + bn*2+1];
    const float nx = out[O_NI + bn*2+0], ny = out[O_NI + bn*2+1];
    const float s  = (float)tid - 31.5f;
    const float x  = cx + s * nx - 0.5f;
    const float y  = cy + s * ny - 0.5f;
    const float fx0 = floorf(x), fy0 = floorf(y);
    sX0[tid] = (int)fx0;  sY0[tid] = (int)fy0;
    sWX[tid] = x - fx0;   sWY[tid] = y - fy0;
  }
  __syncthreads();

  const int l  = tid & (LL_ - 1);
  const int cq = tid >> 6;              // 0..3 (channel sub-group)

  const int x0 = sX0[l], y0 = sY0[l];
  const float wx1 = sWX[l], wy1 = sWY[l];
  const float wx0 = 1.f - wx1, wy0 = 1.f - wy1;
  const bool vx0 = (x0 >= 0) && (x0 < HW_);
  const bool vx1 = (x0 + 1 >= 0) && (x0 + 1 < HW_);
  const bool vy0 = (y0 >= 0) && (y0 < HW_);
  const bool vy1 = (y0 + 1 >= 0) && (y0 + 1 < HW_);
  const float m00 = (vx0 && vy0) ? wx0 * wy0 : 0.f;
  const float m01 = (vx1 && vy0) ? wx1 * wy0 : 0.f;
  const float m10 = (vx0 && vy1) ? wx0 * wy1 : 0.f;
  const float m11 = (vx1 && vy1) ? wx1 * wy1 : 0.f;
  const int x0c = min(max(x0, 0), HW_-1),   x1c = min(max(x0+1, 0), HW_-1);
  const int y0c = min(max(y0, 0), HW_-1),   y1c = min(max(y0+1, 0), HW_-1);
  const int i00 = y0c*HW_ + x0c, i01 = y0c*HW_ + x1c;
  const int i10 = y1c*HW_ + x0c, i11 = y1c*HW_ + x1c;

  const float* fb = feature + (size_t)b * CF_ * PLANE;
#pragma unroll 4
  for (int it = 0; it < 16; ++it) {
    const int c = it * 4 + cq;
    const float* fp = fb + (size_t)c * PLANE;
    if (it < 15) __builtin_prefetch(fp + 4 * PLANE + i00, 0, 1);  // global_prefetch_b8
    const float v = fp[i00] * m00 + fp[i01] * m01
                  + fp[i10] * m10 + fp[i11] * m11;
    out[O_LF + ((size_t)(b * CF_ + c) * NL_ + n) * LL_ + l] = v;
  }
}

// =====================================================================
extern "C" void kernel_launch(void* const* d_in, const int* in_sizes, int n_in,
                              void* d_out, int out_size, void* d_ws, size_t ws_size,
                              hipStream_t stream) {
  (void)in_sizes; (void)n_in; (void)out_size; (void)d_ws; (void)ws_size;
  const float* image   = (const float*)d_in[0];
  const float* feature = (const float*)d_in[1];
  const float* pose    = (const float*)d_in[2];
  const float* camera  = (const float*)d_in[3];
  const float* tv      = (const float*)d_in[4];
  const float* fore    = (const float*)d_in[5];
  const float* back    = (const float*)d_in[6];
  float* out = (float*)d_out;

  k1_pose<<<dim3(13, B_), 32, 0, stream>>>(tv, pose, camera, out);
  k2_hist<<<(B_ * NL_ * LL_ + 255) / 256, 256, 0, stream>>>(image, fore, back, out);
  k3_feat<<<B_ * NL_, 256, 0, stream>>>(feature, out);
}